// GATEncoder_23424751633036
// MI455X (gfx1250) — compile-verified
//
#include <hip/hip_runtime.h>
#include <cstddef>

namespace {
constexpr int N_NODES = 20000;
constexpr int N_EDGES = 320000;
constexpr int G_GRAPHS = 16;
constexpr int D1 = 256;   // H1*C1
constexpr int D2 = 128;   // H2*C2
constexpr float SLOPE = 0.2f;
constexpr float LN_EPS = 1e-5f;
}

typedef float v2f __attribute__((ext_vector_type(2)));
typedef float v8f __attribute__((ext_vector_type(8)));

// ---------- helpers ----------
__device__ __forceinline__ unsigned f2ord(float f) {
  unsigned u = __float_as_uint(f);
  return (u & 0x80000000u) ? ~u : (u | 0x80000000u);
}
__device__ __forceinline__ float ord2f(unsigned o) {
  unsigned u = (o & 0x80000000u) ? (o & 0x7fffffffu) : ~o;
  return __uint_as_float(u);
}
__device__ __forceinline__ float wave_sum(float v) {
#pragma unroll
  for (int off = 16; off > 0; off >>= 1) v += __shfl_xor(v, off, 32);
  return v;
}

// ---------- WMMA GEMM: Out[M,NC] = A[M,K] @ W[K,NC] + bias ----------
// One wave32 per 16x16 output tile; V_WMMA_F32_16X16X4_F32 chained over K with
// FOUR independent accumulators to break the WMMA->WMMA RAW chain (hazard cost).
// Compile-time K/NC -> constant-offset addressing, clean clauses.
// A-tile layout (ISA 7.12.2, 32-bit A 16x4): lanes 0-15 -> K={k,k+1}, lanes 16-31 -> K={k+2,k+3}.
// C/D layout: VGPR r, lanes 0-15 -> M=r, lanes 16-31 -> M=r+8.
template <int K, int NC>
__global__ void gemm_bias_wmma(const float* __restrict__ A, const float* __restrict__ W,
                               const float* __restrict__ bias, float* __restrict__ Out,
                               int M) {
  static_assert(K % 16 == 0, "K must be a multiple of 16");
  const int lane = threadIdx.x & 31;
  const int wave = threadIdx.x >> 5;
  const int tile = blockIdx.x * (blockDim.x >> 5) + wave;
  constexpr int tilesN = NC / 16;
  const int numTiles = (M >> 4) * tilesN;
  if (tile >= numTiles) return;
  const int tm = tile / tilesN;
  const int tn = tile - tm * tilesN;
  const int row0 = tm << 4, n0 = tn << 4;
  const int hi = lane >> 4;     // 0: low half, 1: high half
  const int l15 = lane & 15;
  const float* __restrict__ arow = A + (size_t)(row0 + l15) * K + 2 * hi;  // a.x=arow[k], a.y=arow[k+1]
  const float* __restrict__ wcol = W + (size_t)(2 * hi) * NC + n0 + l15;   // b.x=wcol[k*NC]

  v8f acc0 = {}, acc1 = {}, acc2 = {}, acc3 = {};
#if __has_builtin(__builtin_amdgcn_wmma_f32_16x16x4_f32)
#pragma unroll 4
  for (int k = 0; k < K; k += 16) {
    v2f a0, a1, a2, a3, b0, b1, b2, b3;
    a0.x = arow[k + 0];  a0.y = arow[k + 1];
    a1.x = arow[k + 4];  a1.y = arow[k + 5];
    a2.x = arow[k + 8];  a2.y = arow[k + 9];
    a3.x = arow[k + 12]; a3.y = arow[k + 13];
    b0.x = wcol[(size_t)(k + 0) * NC];  b0.y = wcol[(size_t)(k + 1) * NC];
    b1.x = wcol[(size_t)(k + 4) * NC];  b1.y = wcol[(size_t)(k + 5) * NC];
    b2.x = wcol[(size_t)(k + 8) * NC];  b2.y = wcol[(size_t)(k + 9) * NC];
    b3.x = wcol[(size_t)(k + 12) * NC]; b3.y = wcol[(size_t)(k + 13) * NC];
    acc0 = __builtin_amdgcn_wmma_f32_16x16x4_f32(false, a0, false, b0, (short)0, acc0, false, false);
    acc1 = __builtin_amdgcn_wmma_f32_16x16x4_f32(false, a1, false, b1, (short)0, acc1, false, false);
    acc2 = __builtin_amdgcn_wmma_f32_16x16x4_f32(false, a2, false, b2, (short)0, acc2, false, false);
    acc3 = __builtin_amdgcn_wmma_f32_16x16x4_f32(false, a3, false, b3, (short)0, acc3, false, false);
  }
#else
  for (int k = 0; k < K; ++k) {
    const float bw = W[(size_t)k * NC + n0 + l15];
#pragma unroll
    for (int r = 0; r < 8; ++r)
      acc0[r] = fmaf(A[(size_t)(row0 + r + 8 * hi) * K + k], bw, acc0[r]);
  }
#endif
  const float bv = bias[n0 + l15];
#pragma unroll
  for (int r = 0; r < 8; ++r) {
    const float v = ((acc0[r] + acc1[r]) + (acc2[r] + acc3[r])) + bv;
    Out[(size_t)(row0 + r + 8 * hi) * NC + n0 + l15] = v;
  }
}

// ---------- per-layer init: out=bias, denom=0, emax=-inf(ord 0) ----------
__global__ void init_layer(float* __restrict__ h, const float* __restrict__ bias,
                           float* __restrict__ denom, unsigned* __restrict__ emax,
                           int n, int D, int H) {
  const int t = blockIdx.x * blockDim.x + threadIdx.x;
  if (t < n * H) { denom[t] = 0.f; emax[t] = 0u; }
  if (t < n * D) h[t] = bias[t & (D - 1)];
}

// ---------- fused edge attention scores + segment-max ----------
// One wave per edge. ee = edge_attr @ We computed on the fly from LDS-staged We.
template <int H, int C>
__global__ void edge_scores(const float* __restrict__ xl, const float* __restrict__ xr,
                            const float* __restrict__ eattr, const float* __restrict__ We,
                            const float* __restrict__ att, const int* __restrict__ src,
                            const int* __restrict__ dst, float* __restrict__ score,
                            unsigned* __restrict__ emax, int E) {
  constexpr int D = H * C;
  __shared__ float sWe[8 * D];
  __shared__ float sAtt[D];
  for (int i = threadIdx.x; i < 8 * D; i += blockDim.x) sWe[i] = We[i];
  for (int i = threadIdx.x; i < D; i += blockDim.x) sAtt[i] = att[i];
  __syncthreads();

  const int lane = threadIdx.x & 31;
  const int e = blockIdx.x * (blockDim.x >> 5) + (threadIdx.x >> 5);
  if (e >= E) return;
  const int s = src[e], d = dst[e];
  const float* __restrict__ xls = xl + (size_t)s * D;
  const float* __restrict__ xrd = xr + (size_t)d * D;
  __builtin_prefetch(xls, 0, 0);
  __builtin_prefetch(xrd, 0, 0);
  float ea[8];
#pragma unroll
  for (int k = 0; k < 8; ++k) ea[k] = eattr[(size_t)e * 8 + k];

  float hsum[H];
#pragma unroll
  for (int h = 0; h < H; ++h) hsum[h] = 0.f;
#pragma unroll
  for (int i = 0; i < D / 32; ++i) {
    const int j = lane + 32 * i;
    float ee = 0.f;
#pragma unroll
    for (int k = 0; k < 8; ++k) ee = fmaf(ea[k], sWe[k * D + j], ee);
    const float m = xls[j] + xrd[j] + ee;
    const float lr = m > 0.f ? m : m * SLOPE;
    hsum[(32 * i) / C] = fmaf(lr, sAtt[j], hsum[(32 * i) / C]);  // head index is compile-time
  }
#pragma unroll
  for (int h = 0; h < H; ++h) {
    const float v = wave_sum(hsum[h]);
    if (lane == 0) {
      score[(size_t)e * H + h] = v;
      atomicMax(&emax[(size_t)d * H + h], f2ord(v));  // global_atomic_max_u32 (ordered-float)
    }
  }
}

// ---------- p = exp(e - emax[dst]); denom[dst] += p ----------
template <int H>
__global__ void edge_softmax_p(float* __restrict__ score, const unsigned* __restrict__ emax,
                               float* __restrict__ denom, const int* __restrict__ dst, int E) {
  const int t = blockIdx.x * blockDim.x + threadIdx.x;
  if (t >= E * H) return;
  const int e = t / H, h = t - e * H;
  const int d = dst[e];
  const float mx = ord2f(emax[(size_t)d * H + h]);
  const float p = __expf(score[t] - mx);
  score[t] = p;
  unsafeAtomicAdd(&denom[(size_t)d * H + h], p);  // native global_atomic_add_f32
}

// ---------- out[dst] += (p/denom) * xl[src] ----------
template <int H, int C>
__global__ void edge_aggregate(const float* __restrict__ xl, const float* __restrict__ p,
                               const float* __restrict__ denom, const int* __restrict__ src,
                               const int* __restrict__ dst, float* __restrict__ out, int E) {
  constexpr int D = H * C;
  const int lane = threadIdx.x & 31;
  const int e = blockIdx.x * (blockDim.x >> 5) + (threadIdx.x >> 5);
  if (e >= E) return;
  const int s = src[e], d = dst[e];
  float alpha[H];
#pragma unroll
  for (int h = 0; h < H; ++h)
    alpha[h] = p[(size_t)e * H + h] / (denom[(size_t)d * H + h] + 1e-16f);
  const float* __restrict__ xls = xl + (size_t)s * D;
  float* __restrict__ od = out + (size_t)d * D;
  __builtin_prefetch(xls, 0, 0);
#pragma unroll
  for (int i = 0; i < D / 32; ++i) {
    const int j = lane + 32 * i;
    unsafeAtomicAdd(&od[j], alpha[(32 * i) / C] * xls[j]);
  }
}

// ---------- LayerNorm + ELU, one wave per row, in place ----------
template <int D>
__global__ void ln_elu(float* __restrict__ h, const float* __restrict__ g,
                       const float* __restrict__ b, int n) {
  const int lane = threadIdx.x & 31;
  const int row = blockIdx.x * (blockDim.x >> 5) + (threadIdx.x >> 5);
  if (row >= n) return;
  float* __restrict__ r = h + (size_t)row * D;
  float v[D / 32];
  float s = 0.f;
#pragma unroll
  for (int i = 0; i < D / 32; ++i) { v[i] = r[lane + 32 * i]; s += v[i]; }
  const float mu = wave_sum(s) * (1.f / D);
  float vs = 0.f;
#pragma unroll
  for (int i = 0; i < D / 32; ++i) { const float dlt = v[i] - mu; vs += dlt * dlt; }
  const float rstd = __frsqrt_rn(wave_sum(vs) * (1.f / D) + LN_EPS);
#pragma unroll
  for (int i = 0; i < D / 32; ++i) {
    const int j = lane + 32 * i;
    const float y = (v[i] - mu) * rstd * g[j] + b[j];
    r[j] = y > 0.f ? y : (__expf(y) - 1.f);  // ELU
  }
}

// ---------- global mean pool ----------
__global__ void pool_zero(float* __restrict__ sums, int* __restrict__ cnt) {
  const int t = blockIdx.x * blockDim.x + threadIdx.x;
  if (t < G_GRAPHS * D2) sums[t] = 0.f;
  if (t < G_GRAPHS) cnt[t] = 0;
}
__global__ void pool_accum(const float* __restrict__ h, const int* __restrict__ batch,
                           float* __restrict__ sums, int* __restrict__ cnt, int n) {
  const int lane = threadIdx.x & 31;
  const int row = blockIdx.x * (blockDim.x >> 5) + (threadIdx.x >> 5);
  if (row >= n) return;
  const int g = batch[row];
  const float* __restrict__ r = h + (size_t)row * D2;
#pragma unroll
  for (int i = 0; i < D2 / 32; ++i) {
    const int j = lane + 32 * i;
    unsafeAtomicAdd(&sums[g * D2 + j], r[j]);
  }
  if (lane == 0) atomicAdd(&cnt[g], 1);
}
__global__ void pool_final(const float* __restrict__ sums, const int* __restrict__ cnt,
                           float* __restrict__ out) {
  const int t = blockIdx.x * blockDim.x + threadIdx.x;
  if (t >= G_GRAPHS * D2) return;
  const float c = (float)cnt[t >> 7];
  out[t] = sums[t] / fmaxf(c, 1.f);
}

// ---------- host launch ----------
extern "C" void kernel_launch(void* const* d_in, const int* in_sizes, int n_in,
                              void* d_out, int out_size, void* d_ws, size_t ws_size,
                              hipStream_t stream) {
  const float* x     = (const float*)d_in[0];
  const float* eattr = (const float*)d_in[1];
  const float* W1l = (const float*)d_in[2];  const float* b1l = (const float*)d_in[3];
  const float* W1r = (const float*)d_in[4];  const float* b1r = (const float*)d_in[5];
  const float* W1e = (const float*)d_in[6];  const float* att1 = (const float*)d_in[7];
  const float* bias1 = (const float*)d_in[8];
  const float* W2l = (const float*)d_in[9];  const float* b2l = (const float*)d_in[10];
  const float* W2r = (const float*)d_in[11]; const float* b2r = (const float*)d_in[12];
  const float* W2e = (const float*)d_in[13]; const float* att2 = (const float*)d_in[14];
  const float* bias2 = (const float*)d_in[15];
  const float* g1 = (const float*)d_in[16];  const float* be1 = (const float*)d_in[17];
  const float* g2 = (const float*)d_in[18];  const float* be2 = (const float*)d_in[19];
  const int* eidx  = (const int*)d_in[20];
  const int* batch = (const int*)d_in[21];
  const int* src = eidx;
  const int* dst = eidx + N_EDGES;
  float* out = (float*)d_out;

  // workspace layout (all fits in ~78 MB -> L2 resident)
  char* w = (char*)d_ws;
  float*    xl    = (float*)w;    w += (size_t)N_NODES * D1 * sizeof(float);
  float*    xr    = (float*)w;    w += (size_t)N_NODES * D1 * sizeof(float);
  float*    h1    = (float*)w;    w += (size_t)N_NODES * D1 * sizeof(float);
  float*    h2    = (float*)w;    w += (size_t)N_NODES * D2 * sizeof(float);
  float*    score = (float*)w;    w += (size_t)N_EDGES * 4 * sizeof(float);
  float*    denom = (float*)w;    w += (size_t)N_NODES * 4 * sizeof(float);
  unsigned* emax  = (unsigned*)w; w += (size_t)N_NODES * 4 * sizeof(unsigned);
  float*    psum  = (float*)w;    w += (size_t)G_GRAPHS * D2 * sizeof(float);
  int*      pcnt  = (int*)w;

  const int edgeBlocks = (N_EDGES + 7) / 8;      // 8 waves (edges) per 256-thread block
  const int rowBlocks  = (N_NODES + 7) / 8;

  // ===== layer 1 (H=4, C=64, D=256, K=16) =====
  init_layer<<<(N_NODES * D1 + 255) / 256, 256, 0, stream>>>(h1, bias1, denom, emax, N_NODES, D1, 4);
  {
    const int tiles = (N_NODES / 16) * (D1 / 16);
    gemm_bias_wmma<16, D1><<<(tiles + 3) / 4, 128, 0, stream>>>(x, W1l, b1l, xl, N_NODES);
    gemm_bias_wmma<16, D1><<<(tiles + 3) / 4, 128, 0, stream>>>(x, W1r, b1r, xr, N_NODES);
  }
  edge_scores<4, 64><<<edgeBlocks, 256, 0, stream>>>(xl, xr, eattr, W1e, att1, src, dst, score, emax, N_EDGES);
  edge_softmax_p<4><<<(N_EDGES * 4 + 255) / 256, 256, 0, stream>>>(score, emax, denom, dst, N_EDGES);
  edge_aggregate<4, 64><<<edgeBlocks, 256, 0, stream>>>(xl, score, denom, src, dst, h1, N_EDGES);
  ln_elu<D1><<<rowBlocks, 256, 0, stream>>>(h1, g1, be1, N_NODES);

  // ===== layer 2 (H=1, C=128, D=128, K=256) =====
  init_layer<<<(N_NODES * D2 + 255) / 256, 256, 0, stream>>>(h2, bias2, denom, emax, N_NODES, D2, 1);
  {
    const int tiles = (N_NODES / 16) * (D2 / 16);
    gemm_bias_wmma<D1, D2><<<(tiles + 3) / 4, 128, 0, stream>>>(h1, W2l, b2l, xl, N_NODES);
    gemm_bias_wmma<D1, D2><<<(tiles + 3) / 4, 128, 0, stream>>>(h1, W2r, b2r, xr, N_NODES);
  }
  edge_scores<1, 128><<<edgeBlocks, 256, 0, stream>>>(xl, xr, eattr, W2e, att2, src, dst, score, emax, N_EDGES);
  edge_softmax_p<1><<<(N_EDGES + 255) / 256, 256, 0, stream>>>(score, emax, denom, dst, N_EDGES);
  edge_aggregate<1, 128><<<edgeBlocks, 256, 0, stream>>>(xl, score, denom, src, dst, h2, N_EDGES);
  ln_elu<D2><<<rowBlocks, 256, 0, stream>>>(h2, g2, be2, N_NODES);

  // ===== global mean pool =====
  pool_zero<<<(G_GRAPHS * D2 + 255) / 256, 256, 0, stream>>>(psum, pcnt);
  pool_accum<<<rowBlocks, 256, 0, stream>>>(h2, batch, psum, pcnt, N_NODES);
  pool_final<<<(G_GRAPHS * D2 + 255) / 256, 256, 0, stream>>>(psum, pcnt, out);
}